// ConcatBlock_40252433498680
// MI455X (gfx1250) — compile-verified
//
#include <hip/hip_runtime.h>

// Fused: z = cat(x_bthc, s_bc) @ W^T + bias; z = LayerNorm(PReLU(z)); y = x + z
// B=4, C1=512, T=256, H=64, AUX=256, OUT=512, K=768.
// One workgroup (512 thr = 16 waves, wave32) per (b,t): 64 rows x 512 cols.
// fp32 WMMA 16x16x4 keeps reference numerics; ~270MB HBM @ 23.3TB/s is
// balanced against the fp32 matrix pipe.
//  - Each wave owns only 8 C-tiles (64 acc VGPRs) so the scheduler can keep
//    all 8 B fragments of a k-step in flight (batched ds_load_b64s, one wait,
//    then 8 back-to-back v_wmma) instead of serializing load->wait->wmma.
//  - LDS holds A/B as k-PAIRS (float2): every fragment is ONE aligned 8B DS
//    load into an even VGPR pair; pair-row strides (80/528 float2) put lanes
//    0-15 and 16-31 in disjoint bank halves (conflict-free).
//  - Next K-chunk globals are prefetched into registers during the WMMAs.

typedef float v2f __attribute__((ext_vector_type(2)));
typedef float v8f __attribute__((ext_vector_type(8)));

#define BB   4
#define C1   512
#define TT   256
#define HH   64
#define AUXD 256
#define KDIM 768
#define NOUT 512
#define KC   32     // K chunk staged in LDS per iteration (16 k-pairs)

// LDS float offsets (one dynamic-shared allocation):
//  GEMM:  A  [0, 16*160)             16 k-pair rows x 64 cols (float2 row=160f)
//         B  [2560, 2560+16*1056)    16 k-pair rows x 512 cols (float2 row=1056f)
//  Epilogue: y stage [0, 64*513)     64 rows x 512 cols, stride 513 (pad)
//  Partials: [32832, 32832+512)      64 rows x {sum[4], sq[4]}
#define A_ROWF   160
#define B_ROWF   1056
#define LDS_A    0
#define LDS_B    (16 * A_ROWF)         // 2560
#define Y_STRIDE 513
#define LDS_PART (HH * Y_STRIDE)       // 32832
#define LDS_FLOATS (LDS_PART + HH * 8) // 33344 floats = 133376 B

__global__ __launch_bounds__(512)
void fused_concat_linear_ln_kernel(
    const float* __restrict__ x,     // (B, C1, T, H)
    const float* __restrict__ s,     // (B, AUX)
    const float* __restrict__ Wm,    // (OUT, C1+AUX)
    const float* __restrict__ bias,  // (OUT,)
    const float* __restrict__ prelu_a,
    const float* __restrict__ lnw,   // (OUT,)
    const float* __restrict__ lnb,   // (OUT,)
    float* __restrict__ out)         // (B, OUT, T, H)
{
  extern __shared__ float smem[];

  const int tid  = threadIdx.x;       // 0..511
  const int bt   = blockIdx.x;
  const int b    = bt >> 8;           // grid = B*T, T=256
  const int t    = bt & 255;

  const int lane  = tid & 31;
  const int wave  = tid >> 5;         // 0..15
  const int lo    = lane & 15;        // N / M index within fragment
  const int hi    = lane >> 4;        // lanes16-31: k-pair +1 (A/B), row +8 (C)
  const int mtile = wave & 3;         // 4 tiles of 16 rows
  const int nq    = wave >> 2;        // 4 quarters of 128 cols
  const int nb    = nq * 128;
  const int m     = mtile * 16 + lo;

  v8f acc[8];
  #pragma unroll
  for (int i = 0; i < 8; ++i)
    #pragma unroll
    for (int e = 0; e < 8; ++e) acc[i][e] = 0.0f;

  // staging thread coords (512 threads)
  const int h    = tid & 63;          // A: lane-contiguous h -> coalesced x
  const int cp0  = tid >> 6;          // A: base k-pair (0..7), pairs cp0 + 8*i
  const int o0   = tid & 127;         // B: lane-contiguous o
  const int cgrp = tid >> 7;          // B: 0..3 -> c4 = cgrp*4 + 16*j

  float2 aReg[2];
  float4 bReg[2][4];

  // prefetch one K-chunk's globals into registers
  auto prefetch = [&](int kb) {
    #pragma unroll
    for (int i = 0; i < 2; ++i) {
      const int p  = cp0 + i * 8;               // k-pair 0..15
      const int c0 = kb + 2 * p;
      const int c1 = c0 + 1;
      float v0, v1;
      if (c0 < C1) v0 = x[(((size_t)b * C1 + c0) * TT + t) * HH + h];
      else         v0 = s[(size_t)b * AUXD + (c0 - C1)];
      if (c1 < C1) v1 = x[(((size_t)b * C1 + c1) * TT + t) * HH + h];
      else         v1 = s[(size_t)b * AUXD + (c1 - C1)];
      aReg[i] = make_float2(v0, v1);
    }
    #pragma unroll
    for (int j = 0; j < 2; ++j) {
      const int c4 = cgrp * 4 + 16 * j;
      #pragma unroll
      for (int i = 0; i < 4; ++i) {
        const int o = o0 + 128 * i;
        bReg[j][i] = *(const float4*)(&Wm[(size_t)o * KDIM + kb + c4]); // L2-hot
      }
    }
  };

  // write prefetched registers into LDS (k-pair float2 layout, b64 stores)
  auto stage = [&]() {
    #pragma unroll
    for (int i = 0; i < 2; ++i) {
      const int p = cp0 + i * 8;
      *(float2*)(&smem[LDS_A + p * A_ROWF + 2 * h]) = aReg[i];
    }
    #pragma unroll
    for (int j = 0; j < 2; ++j) {
      const int jp = (cgrp * 4 + 16 * j) >> 1;  // k-pair row
      #pragma unroll
      for (int i = 0; i < 4; ++i) {
        const int o = o0 + 128 * i;
        *(float2*)(&smem[LDS_B + (jp + 0) * B_ROWF + 2 * o]) =
            make_float2(bReg[j][i].x, bReg[j][i].y);
        *(float2*)(&smem[LDS_B + (jp + 1) * B_ROWF + 2 * o]) =
            make_float2(bReg[j][i].z, bReg[j][i].w);
      }
    }
  };

  prefetch(0);
  for (int kb = 0; kb < KDIM; kb += KC) {
    stage();
    __syncthreads();
    if (kb + KC < KDIM) prefetch(kb + KC);      // overlap with WMMA below

    #pragma unroll
    for (int kk = 0; kk < KC; kk += 4) {
      const int jrow = (kk >> 1) + hi;          // k-pair row for this fragment
      const v2f a = *(const v2f*)(&smem[LDS_A + jrow * A_ROWF + 2 * m]);
      v2f bf[8];                                 // batch loads, then 8 WMMAs
      #pragma unroll
      for (int nt = 0; nt < 8; ++nt) {
        const int n = nb + nt * 16 + lo;
        bf[nt] = *(const v2f*)(&smem[LDS_B + jrow * B_ROWF + 2 * n]);
      }
      #pragma unroll
      for (int nt = 0; nt < 8; ++nt) {
        acc[nt] = __builtin_amdgcn_wmma_f32_16x16x4_f32(
            false, a, false, bf[nt], (short)0, acc[nt], false, false);
      }
    }
    __syncthreads();
  }

  // ---- epilogue: bias + PReLU + LayerNorm(512) per row -------------------
  const float alpha = prelu_a[0];

  float s1[8], s2[8];
  #pragma unroll
  for (int v = 0; v < 8; ++v) { s1[v] = 0.0f; s2[v] = 0.0f; }

  #pragma unroll
  for (int nt = 0; nt < 8; ++nt) {
    const int n = nb + nt * 16 + lo;
    const float bv = bias[n];
    #pragma unroll
    for (int v = 0; v < 8; ++v) {
      float z = acc[nt][v] + bv;
      acc[nt][v] = z;                        // keep z; re-apply PReLU later
      float p = (z >= 0.0f) ? z : alpha * z;
      s1[v] += p;
      s2[v] += p * p;
    }
  }
  // reduce across the 16 lanes that share each row (wave32, width-16 xor)
  #pragma unroll
  for (int v = 0; v < 8; ++v) {
    #pragma unroll
    for (int off = 1; off < 16; off <<= 1) {
      s1[v] += __shfl_xor(s1[v], off, 16);
      s2[v] += __shfl_xor(s2[v], off, 16);
    }
  }
  // exchange the four column-quarters of each row through LDS
  if (lo == 0) {
    #pragma unroll
    for (int v = 0; v < 8; ++v) {
      const int row = mtile * 16 + hi * 8 + v;   // C layout: M = v + 8*hi
      smem[LDS_PART + row * 8 + nq]     = s1[v];
      smem[LDS_PART + row * 8 + 4 + nq] = s2[v];
    }
  }
  __syncthreads();

  float mu[8], rs[8];
  #pragma unroll
  for (int v = 0; v < 8; ++v) {
    const int row = mtile * 16 + hi * 8 + v;
    float su = 0.0f, sq = 0.0f;
    #pragma unroll
    for (int q = 0; q < 4; ++q) {
      su += smem[LDS_PART + row * 8 + q];
      sq += smem[LDS_PART + row * 8 + 4 + q];
    }
    const float mean = su * (1.0f / NOUT);
    const float var  = sq * (1.0f / NOUT) - mean * mean;
    mu[v] = mean;
    rs[v] = rsqrtf(var + 1e-8f);
  }

  // normalize and transpose-stage into LDS (row stride 513 -> conflict-free
  // strided reads at store time)
  #pragma unroll
  for (int nt = 0; nt < 8; ++nt) {
    const int n = nb + nt * 16 + lo;
    const float gw = lnw[n];
    const float gb = lnb[n];
    #pragma unroll
    for (int v = 0; v < 8; ++v) {
      const int row = mtile * 16 + hi * 8 + v;
      const float z = acc[nt][v];
      const float p = (z >= 0.0f) ? z : alpha * z;
      smem[row * Y_STRIDE + n] = (p - mu[v]) * rs[v] * gw + gb;
    }
  }
  __syncthreads();

  // ---- residual add + coalesced store along h (float4) -------------------
  const int h4 = (tid & 15) * 4;  // 4 consecutive h per thread
  const int og = tid >> 4;        // 32 o per pass
  for (int o = og; o < NOUT; o += 32) {
    float4 r;
    r.x = smem[(h4 + 0) * Y_STRIDE + o];
    r.y = smem[(h4 + 1) * Y_STRIDE + o];
    r.z = smem[(h4 + 2) * Y_STRIDE + o];
    r.w = smem[(h4 + 3) * Y_STRIDE + o];
    const size_t gi = (((size_t)b * C1 + o) * TT + t) * HH + h4;
    const float4 xv = *(const float4*)(&x[gi]);   // residual (L2-hot)
    r.x += xv.x; r.y += xv.y; r.z += xv.z; r.w += xv.w;
    *(float4*)(&out[gi]) = r;
  }
}

extern "C" void kernel_launch(void* const* d_in, const int* in_sizes, int n_in,
                              void* d_out, int out_size, void* d_ws, size_t ws_size,
                              hipStream_t stream) {
  const float* x    = (const float*)d_in[0];
  const float* s    = (const float*)d_in[1];
  // d_in[2..4] = prelu1_a, ln1_w, ln1_b -> dead code in the reference
  const float* Wm   = (const float*)d_in[5];
  const float* bias = (const float*)d_in[6];
  const float* a2   = (const float*)d_in[7];
  const float* lnw  = (const float*)d_in[8];
  const float* lnb  = (const float*)d_in[9];
  float* out = (float*)d_out;

  const size_t shmem = (size_t)LDS_FLOATS * sizeof(float);  // ~130.2 KB
  fused_concat_linear_ln_kernel<<<dim3(BB * TT), dim3(512), shmem, stream>>>(
      x, s, Wm, bias, a2, lnw, lnb, out);
}